// Qwen2_5MoEExpertRouter_47038481826556
// MI455X (gfx1250) — compile-verified
//
#include <hip/hip_runtime.h>
#include <hip/hip_bf16.h>
#include <math.h>

typedef float v2f __attribute__((ext_vector_type(2)));
typedef float v8f __attribute__((ext_vector_type(8)));

#define DDIM 2048
#define EXP  64
#define KC   128                 // K chunk staged in LDS
#define LDSS (KC + 4)            // padded row stride (floats) -> conflict-free ds_load_b64

// ---------------------------------------------------------------------------
// Kernel 1: router GEMM, logits[N,64] = x[N,2048] * w[64,2048]^T via
// V_WMMA_F32_16X16X4_F32. 4 waves/WG, 16 tokens per wave, all 64 experts.
// ---------------------------------------------------------------------------
__global__ __launch_bounds__(128) void router_gemm(const float* __restrict__ x,
                                                   const float* __restrict__ w,
                                                   float* __restrict__ logits) {
    __shared__ float wlds[EXP * LDSS];           // 64*132*4 = 33.8 KB

    const int tid  = threadIdx.x;
    const int lane = tid & 31;
    const int wave = tid >> 5;                   // 0..3
    const int m    = lane & 15;                  // row/col within 16x16 tile
    const int half = lane >> 4;                  // K half: 0 -> K{0,1}, 1 -> K{2,3}
    const int wgTok = blockIdx.x * 64;           // 64 tokens per workgroup
    const int row   = wgTok + wave * 16 + m;     // A-fragment token row

    const float* xrow = x + (size_t)row * DDIM + 2 * half;

    v8f acc0 = {}, acc1 = {}, acc2 = {}, acc3 = {};

    for (int c = 0; c < DDIM / KC; ++c) {
        __syncthreads();                          // protect LDS reuse
        // cooperatively stage w[:, c*KC .. c*KC+KC) into LDS (b128 loads/stores)
        #pragma unroll
        for (int it = 0; it < (EXP * KC) / (128 * 4); ++it) {   // 16 iters
            int f  = (it * 128 + tid) * 4;        // flat float index
            int e  = f >> 7;                      // f / KC
            int kl = f & (KC - 1);
            const float4 v = *(const float4*)(w + (size_t)e * DDIM + c * KC + kl);
            *(float4*)(&wlds[e * LDSS + kl]) = v;
        }
        __syncthreads();

        const float* xk = xrow + c * KC;
        #pragma unroll 4
        for (int kk = 0; kk < KC; kk += 4) {
            v2f a = *(const v2f*)(xk + kk);       // A: x[row][k+2*half .. +1]
            const int ko = kk + 2 * half;
            v2f b0 = *(const v2f*)(&wlds[( 0 + m) * LDSS + ko]);
            v2f b1 = *(const v2f*)(&wlds[(16 + m) * LDSS + ko]);
            v2f b2 = *(const v2f*)(&wlds[(32 + m) * LDSS + ko]);
            v2f b3 = *(const v2f*)(&wlds[(48 + m) * LDSS + ko]);
            acc0 = __builtin_amdgcn_wmma_f32_16x16x4_f32(false, a, false, b0,
                                                         (short)0, acc0, false, false);
            acc1 = __builtin_amdgcn_wmma_f32_16x16x4_f32(false, a, false, b1,
                                                         (short)0, acc1, false, false);
            acc2 = __builtin_amdgcn_wmma_f32_16x16x4_f32(false, a, false, b2,
                                                         (short)0, acc2, false, false);
            acc3 = __builtin_amdgcn_wmma_f32_16x16x4_f32(false, a, false, b3,
                                                         (short)0, acc3, false, false);
        }
    }

    // C/D layout: VGPR r -> M = r + 8*half, N = lane&15
    #pragma unroll
    for (int r = 0; r < 8; ++r) {
        const size_t tok = (size_t)(wgTok + wave * 16 + r + 8 * half);
        float* o = logits + tok * EXP + m;
        o[ 0] = acc0[r];
        o[16] = acc1[r];
        o[32] = acc2[r];
        o[48] = acc3[r];
    }
}

// ---------------------------------------------------------------------------
// Kernel 2: per-token softmax, top-2, renorm, dispatch/combine scatter.
// One thread per token (64 logits held in registers).
// ---------------------------------------------------------------------------
__global__ __launch_bounds__(256) void softmax_topk(const float* __restrict__ logits,
                                                    float* __restrict__ dispatch,
                                                    float* __restrict__ combine,
                                                    float* __restrict__ probs,
                                                    float* __restrict__ tk_idx,
                                                    float* __restrict__ tk_p) {
    const int t = blockIdx.x * blockDim.x + threadIdx.x;
    const float* lrow = logits + (size_t)t * EXP;

    float l[EXP];
    float mx = -INFINITY;
    #pragma unroll
    for (int e = 0; e < EXP; e += 4) {
        float4 v = *(const float4*)(lrow + e);
        l[e] = v.x; l[e + 1] = v.y; l[e + 2] = v.z; l[e + 3] = v.w;
        mx = fmaxf(mx, fmaxf(fmaxf(v.x, v.y), fmaxf(v.z, v.w)));
    }

    float sum = 0.0f;
    float v0 = -INFINITY, v1 = -INFINITY;
    int   i0 = 0, i1 = 0;
    #pragma unroll
    for (int e = 0; e < EXP; ++e) {
        float p = expf(l[e] - mx);
        l[e] = p;
        sum += p;
        if (p > v0)      { v1 = v0; i1 = i0; v0 = p; i0 = e; }   // strict >: lowest idx wins ties
        else if (p > v1) { v1 = p;  i1 = e; }
    }

    const float rinv = 1.0f / sum;
    const float rtk  = 1.0f / (v0 + v1);
    const float pn0  = v0 * rtk;
    const float pn1  = v1 * rtk;

    float* pr = probs    + (size_t)t * EXP;
    float* dp = dispatch + (size_t)t * EXP;
    float* cb = combine  + (size_t)t * EXP;
    #pragma unroll
    for (int e = 0; e < EXP; e += 4) {
        float4 p4 = make_float4(l[e] * rinv, l[e + 1] * rinv,
                                l[e + 2] * rinv, l[e + 3] * rinv);
        *(float4*)(pr + e) = p4;
        float4 d4;
        d4.x = (e     == i0) ? pn0 : ((e     == i1) ? pn1 : 0.0f);
        d4.y = (e + 1 == i0) ? pn0 : ((e + 1 == i1) ? pn1 : 0.0f);
        d4.z = (e + 2 == i0) ? pn0 : ((e + 2 == i1) ? pn1 : 0.0f);
        d4.w = (e + 3 == i0) ? pn0 : ((e + 3 == i1) ? pn1 : 0.0f);
        *(float4*)(dp + e) = d4;
        *(float4*)(cb + e) = d4;
    }

    tk_idx[(size_t)t * 2]     = (float)i0;
    tk_idx[(size_t)t * 2 + 1] = (float)i1;
    tk_p  [(size_t)t * 2]     = pn0;
    tk_p  [(size_t)t * 2 + 1] = pn1;
}

// ---------------------------------------------------------------------------
// Kernel 3: per-expert (mean over tokens of prob)^2, fixed-order tree reduce.
// One block per expert -> deterministic.
// ---------------------------------------------------------------------------
__global__ __launch_bounds__(256) void expert_meansq(const float* __restrict__ probs,
                                                     float* __restrict__ wsf,
                                                     int ntok) {
    __shared__ float red[256];
    const int e = blockIdx.x;
    const int t = threadIdx.x;
    float s = 0.0f;
    for (int j = t; j < ntok; j += 256)
        s += probs[(size_t)j * EXP + e];
    red[t] = s;
    __syncthreads();
    for (int off = 128; off > 0; off >>= 1) {
        if (t < off) red[t] += red[t + off];
        __syncthreads();
    }
    if (t == 0) {
        float p = red[0] / (float)ntok;
        wsf[e] = p * p;
    }
}

// Kernel 4: aux_loss = mean_e(E * p_e^2) = sum_e p_e^2
__global__ __launch_bounds__(64) void aux_reduce(const float* __restrict__ wsf,
                                                 float* __restrict__ out_aux) {
    __shared__ float red[64];
    const int t = threadIdx.x;
    red[t] = wsf[t];
    __syncthreads();
    for (int off = 32; off > 0; off >>= 1) {
        if (t < off) red[t] += red[t + off];
        __syncthreads();
    }
    if (t == 0) *out_aux = red[0];
}

// ---------------------------------------------------------------------------
extern "C" void kernel_launch(void* const* d_in, const int* in_sizes, int n_in,
                              void* d_out, int out_size, void* d_ws, size_t ws_size,
                              hipStream_t stream) {
    const float* x = (const float*)d_in[0];   // [4,4096,2048] f32
    const float* w = (const float*)d_in[1];   // [64,2048]     f32

    const int N = in_sizes[0] / DDIM;         // 16384 tokens
    const size_t NE = (size_t)N * EXP;

    float* out      = (float*)d_out;
    float* dispatch = out;                    // [N,E]
    float* combine  = out + NE;               // [N,E]
    float* logits   = out + 2 * NE;           // [N,E]
    float* probs    = out + 3 * NE;           // [N,E]
    float* aux      = out + 4 * NE;           // [1]
    float* tk_idx   = aux + 1;                // [N,2] (indices as floats)
    float* tk_p     = tk_idx + (size_t)N * 2; // [N,2]

    float* wsf = (float*)d_ws;                // 64 floats scratch

    router_gemm <<<N / 64, 128, 0, stream>>>(x, w, logits);
    softmax_topk<<<N / 256, 256, 0, stream>>>(logits, dispatch, combine, probs,
                                              tk_idx, tk_p);
    expert_meansq<<<EXP, 256, 0, stream>>>(probs, wsf, N);
    aux_reduce  <<<1, 64, 0, stream>>>(wsf, aux);
}